// DHGAT_Layer_45569603011138
// MI455X (gfx1250) — compile-verified
//
#include <hip/hip_runtime.h>
#include <hip/hip_bf16.h>
#include <cmath>
#include <cstdint>

// ---------------------------------------------------------------------------
// DHGAT layer for MI455X (gfx1250), wave32.
//  - Dense projections x@W via V_WMMA_F32_16X16X4_F32 (exact fp32 tensor op).
//  - Edge scatter-softmax passes: wave-per-edge, float4/lane, __shfl_xor
//    reduction (warpSize==32), f32 global atomics (native on CDNA5).
//  - Gumbel hard gate is exactly one-hot in forward -> gate-skip edges.
// ---------------------------------------------------------------------------

#define NEG_SLOPE 0.2f
#define INV_TAU   2.0f   // 1 / 0.5

typedef float v2f __attribute__((ext_vector_type(2)));
typedef float v8f __attribute__((ext_vector_type(8)));

static __device__ __forceinline__ float lrelu(float v) {
  return v > 0.0f ? v : NEG_SLOPE * v;
}

// float atomic max via sign-aware integer atomics (init to -inf).
static __device__ __forceinline__ void atomicMaxF32(float* addr, float val) {
  if (val >= 0.0f)
    atomicMax((int*)addr, __float_as_int(val));
  else
    atomicMin((unsigned int*)addr, __float_as_uint(val));
}

// ------------------------------- utility -----------------------------------

__global__ void fill_f32(float* __restrict__ p, float v, size_t n) {
  size_t i = (size_t)blockIdx.x * blockDim.x + threadIdx.x;
  if (i < n) p[i] = v;
}

__global__ void add_bias128(float* __restrict__ out,
                            const float* __restrict__ bias, size_t n) {
  size_t i = (size_t)blockIdx.x * blockDim.x + threadIdx.x;
  if (i < n) out[i] += bias[i & 127];
}

// -------------------- dense projection: C = A(128) @ B(128x128) + bias -----
// One wave per 16x16 output tile; K=128 as 32 chained f32 WMMAs.
// f32 WMMA fragment layouts per CDNA5 ISA 7.12.2:
//   A 16x4 : lane<16 row=lane, lane>=16 row=lane-16; VGPR v holds K = 2h+v
//   B 4x16 : VGPR v holds row K = 2h+v, col = lane&15
//   C 16x16: VGPR i holds row (i + 8h), col = lane&15
__global__ __launch_bounds__(32) void gemm128_wmma(
    const float* __restrict__ A, const float* __restrict__ B,
    const float* __restrict__ bias, float* __restrict__ C, int nrows) {
  const int rowBase = blockIdx.x << 4;
  const int colBase = blockIdx.y << 4;
  const int lane = threadIdx.x;
  const int h = lane >> 4;     // lane half (0/1)
  const int l = lane & 15;

  int r = rowBase + l;
  if (r >= nrows) r = nrows - 1;            // clamped load (EXEC stays all-1s)
  const float* arow = A + (size_t)r * 128;

  v8f acc = {};
#pragma unroll
  for (int k = 0; k < 128; k += 4) {
    v2f a, b;
    a.x = arow[k + 2 * h];
    a.y = arow[k + 2 * h + 1];
    b.x = B[(size_t)(k + 2 * h) * 128 + colBase + l];
    b.y = B[(size_t)(k + 2 * h + 1) * 128 + colBase + l];
    acc = __builtin_amdgcn_wmma_f32_16x16x4_f32(
        /*neg_a=*/false, a, /*neg_b=*/false, b,
        /*c_mod=*/(short)0, acc, /*reuse_a=*/false, /*reuse_b=*/false);
  }

  const float bc = bias[colBase + l];
#pragma unroll
  for (int i = 0; i < 8; ++i) {
    int ro = rowBase + i + 8 * h;
    if (ro < nrows) C[(size_t)ro * 128 + colBase + l] = acc[i] + bc;
  }
}

// ---------------- decision projections (out dim = 2, scalar) ---------------

__global__ void dec_proj(const float* __restrict__ x,
                         const float* __restrict__ Wl, const float* __restrict__ bl,
                         const float* __restrict__ Wr, const float* __restrict__ br,
                         float* __restrict__ xld, float* __restrict__ xrd, int n) {
  int i = blockIdx.x * blockDim.x + threadIdx.x;
  if (i >= n) return;
  const float* xi = x + (size_t)i * 128;
  float a0 = 0.f, a1 = 0.f, b0 = 0.f, b1 = 0.f;
#pragma unroll 4
  for (int k = 0; k < 128; ++k) {
    float v = xi[k];
    a0 += v * Wl[2 * k];     a1 += v * Wl[2 * k + 1];
    b0 += v * Wr[2 * k];     b1 += v * Wr[2 * k + 1];
  }
  xld[2 * i] = a0 + bl[0];   xld[2 * i + 1] = a1 + bl[1];
  xrd[2 * i] = b0 + br[0];   xrd[2 * i + 1] = b1 + br[1];
}

// ---------------------- decision GAT edge passes (D=2) ---------------------

static __device__ __forceinline__ void edge_sd(const long long* e, long long E,
                                               long long t, int& s, int& d) {
  if (t < E) { s = (int)e[t]; d = (int)e[E + t]; }
  else       { s = (int)(t - E); d = s; }          // self-loop
}

__global__ void dec_score(const long long* __restrict__ eidx, long long E,
                          long long T, const float* __restrict__ xld,
                          const float* __restrict__ xrd,
                          const float* __restrict__ att,
                          float* __restrict__ escr, float* __restrict__ m) {
  long long t = (long long)blockIdx.x * blockDim.x + threadIdx.x;
  if (t >= T) return;
  int s, d; edge_sd(eidx, E, t, s, d);
  float e = lrelu(xld[2 * s]     + xrd[2 * d])     * att[0] +
            lrelu(xld[2 * s + 1] + xrd[2 * d + 1]) * att[1];
  escr[t] = e;
  atomicMaxF32(&m[d], e);
}

__global__ void dec_norm(const long long* __restrict__ eidx, long long E,
                         long long T, const float* __restrict__ escr,
                         const float* __restrict__ m,
                         float* __restrict__ ascr, float* __restrict__ sden) {
  long long t = (long long)blockIdx.x * blockDim.x + threadIdx.x;
  if (t >= T) return;
  int s, d; edge_sd(eidx, E, t, s, d); (void)s;
  float a = __expf(escr[t] - m[d]);
  ascr[t] = a;
  atomicAdd(&sden[d], a);
}

__global__ void dec_agg(const long long* __restrict__ eidx, long long E,
                        long long T, const float* __restrict__ ascr,
                        const float* __restrict__ sden,
                        const float* __restrict__ xld, float* __restrict__ acc) {
  long long t = (long long)blockIdx.x * blockDim.x + threadIdx.x;
  if (t >= T) return;
  int s, d; edge_sd(eidx, E, t, s, d);
  float alpha = ascr[t] / (sden[d] + 1e-16f);
  atomicAdd(&acc[2 * d],     alpha * xld[2 * s]);
  atomicAdd(&acc[2 * d + 1], alpha * xld[2 * s + 1]);
}

// hard gumbel-softmax forward == one-hot(argmax((logits+g)/tau))
__global__ void decide(const float* __restrict__ acc,
                       const float* __restrict__ bias_d,
                       const float* __restrict__ gumbel,
                       float* __restrict__ gates, int n) {
  int i = blockIdx.x * blockDim.x + threadIdx.x;
  if (i >= n) return;
  float l0 = (acc[2 * i]     + bias_d[0] + gumbel[2 * i])     * INV_TAU;
  float l1 = (acc[2 * i + 1] + bias_d[1] + gumbel[2 * i + 1]) * INV_TAU;
  int idx = (l1 > l0) ? 1 : 0;                 // jnp.argmax: first on tie
  gates[2 * i]     = (idx == 0) ? 1.0f : 0.0f;
  gates[2 * i + 1] = (idx == 1) ? 1.0f : 0.0f;
}

// ---------------------- shared GAT edge passes (D=128) ---------------------
// wave-per-edge: each of 32 lanes handles one float4 channel chunk.

__global__ __launch_bounds__(256) void gat_score(
    const long long* __restrict__ eidx, long long E, long long T,
    const float* __restrict__ xl, const float* __restrict__ xr,
    const float* __restrict__ att, const float* __restrict__ gates, int which,
    float* __restrict__ escr, float* __restrict__ m) {
  long long w = (long long)blockIdx.x * (blockDim.x >> 5) + (threadIdx.x >> 5);
  if (w >= T) return;
  const int lane = threadIdx.x & 31;
  int s, d; edge_sd(eidx, E, w, s, d);
  if (gates[2 * d + which] == 0.0f) return;    // dst node gated off
  const float4 vl = ((const float4*)(xl + (size_t)s * 128))[lane];
  const float4 vr = ((const float4*)(xr + (size_t)d * 128))[lane];
  const float4 va = ((const float4*)att)[lane];
  float p = lrelu(vl.x + vr.x) * va.x + lrelu(vl.y + vr.y) * va.y +
            lrelu(vl.z + vr.z) * va.z + lrelu(vl.w + vr.w) * va.w;
#pragma unroll
  for (int off = 16; off; off >>= 1) p += __shfl_xor(p, off);
  if (lane == 0) {
    escr[w] = p;
    atomicMaxF32(&m[d], p);
  }
}

__global__ void gat_norm(const long long* __restrict__ eidx, long long E,
                         long long T, const float* __restrict__ escr,
                         const float* __restrict__ m,
                         const float* __restrict__ gates, int which,
                         float* __restrict__ ascr, float* __restrict__ sden) {
  long long t = (long long)blockIdx.x * blockDim.x + threadIdx.x;
  if (t >= T) return;
  int s, d; edge_sd(eidx, E, t, s, d); (void)s;
  if (gates[2 * d + which] == 0.0f) return;
  float a = __expf(escr[t] - m[d]);
  ascr[t] = a;
  atomicAdd(&sden[d], a);
}

__global__ __launch_bounds__(256) void gat_agg(
    const long long* __restrict__ eidx, long long E, long long T,
    const float* __restrict__ ascr, const float* __restrict__ sden,
    const float* __restrict__ xl, const float* __restrict__ gates, int which,
    float* __restrict__ out) {
  long long w = (long long)blockIdx.x * (blockDim.x >> 5) + (threadIdx.x >> 5);
  if (w >= T) return;
  const int lane = threadIdx.x & 31;
  int s, d; edge_sd(eidx, E, w, s, d);
  if (gates[2 * d + which] == 0.0f) return;    // gate==1 on active -> no mult
  float alpha = ascr[w] / (sden[d] + 1e-16f);
  const float4 vl = ((const float4*)(xl + (size_t)s * 128))[lane];
  float* po = out + (size_t)d * 128 + lane * 4;
  atomicAdd(po + 0, alpha * vl.x);
  atomicAdd(po + 1, alpha * vl.y);
  atomicAdd(po + 2, alpha * vl.z);
  atomicAdd(po + 3, alpha * vl.w);
}

// ---------------------------------------------------------------------------

static inline unsigned cdiv(long long a, long long b) {
  return (unsigned)((a + b - 1) / b);
}

extern "C" void kernel_launch(void* const* d_in, const int* in_sizes, int n_in,
                              void* d_out, int out_size, void* d_ws, size_t ws_size,
                              hipStream_t stream) {
  const float*     x      = (const float*)d_in[0];
  const long long* e_dec  = (const long long*)d_in[1];   // jnp.int64
  const long long* e_0    = (const long long*)d_in[2];
  const long long* e_1    = (const long long*)d_in[3];
  const float*     gumbel = (const float*)d_in[4];
  const float*     Wl_g   = (const float*)d_in[5];
  const float*     Wr_g   = (const float*)d_in[6];
  const float*     bl_g   = (const float*)d_in[7];
  const float*     br_g   = (const float*)d_in[8];
  const float*     att_g  = (const float*)d_in[9];
  const float*     bias_g = (const float*)d_in[10];
  const float*     Wl_d   = (const float*)d_in[11];
  const float*     Wr_d   = (const float*)d_in[12];
  const float*     bl_d   = (const float*)d_in[13];
  const float*     br_d   = (const float*)d_in[14];
  const float*     att_d  = (const float*)d_in[15];
  const float*     bias_d = (const float*)d_in[16];

  const int       N = in_sizes[0] / 128;
  const long long E = in_sizes[1] / 2;
  const long long T = E + (long long)N;      // edges + self loops

  float* out = (float*)d_out;                // N x 128 fp32

  // ---- workspace carve-up -------------------------------------------------
  char* ws = (char*)d_ws;
  float* xl_g  = (float*)ws; ws += (size_t)N * 128 * 4;
  float* xr_g  = (float*)ws; ws += (size_t)N * 128 * 4;
  float* xl_d  = (float*)ws; ws += (size_t)N * 2 * 4;
  float* xr_d  = (float*)ws; ws += (size_t)N * 2 * 4;
  float* lacc  = (float*)ws; ws += (size_t)N * 2 * 4;   // decision logits acc
  float* gates = (float*)ws; ws += (size_t)N * 2 * 4;   // one-hot gates
  float* mbuf  = (float*)ws; ws += (size_t)N * 4;       // segment max
  float* sbuf  = (float*)ws; ws += (size_t)N * 4;       // segment sum
  float* escr  = (float*)ws; ws += (size_t)T * 4;       // per-edge score
  float* ascr  = (float*)ws; ws += (size_t)T * 4;       // per-edge exp

  const size_t NO = (size_t)N * 128;
  const float NEG_INF = -INFINITY;

  // ---- init ---------------------------------------------------------------
  fill_f32<<<cdiv(NO, 256), 256, 0, stream>>>(out, 0.0f, NO);
  fill_f32<<<cdiv(2 * N, 256), 256, 0, stream>>>(lacc, 0.0f, (size_t)2 * N);

  // ---- dense projections (WMMA f32) --------------------------------------
  dim3 gg(cdiv(N, 16), 8);
  gemm128_wmma<<<gg, 32, 0, stream>>>(x, Wl_g, bl_g, xl_g, N);
  gemm128_wmma<<<gg, 32, 0, stream>>>(x, Wr_g, br_g, xr_g, N);
  dec_proj<<<cdiv(N, 256), 256, 0, stream>>>(x, Wl_d, bl_d, Wr_d, br_d,
                                             xl_d, xr_d, N);

  // ---- decision GAT -------------------------------------------------------
  fill_f32<<<cdiv(N, 256), 256, 0, stream>>>(mbuf, NEG_INF, (size_t)N);
  fill_f32<<<cdiv(N, 256), 256, 0, stream>>>(sbuf, 0.0f, (size_t)N);
  dec_score<<<cdiv(T, 256), 256, 0, stream>>>(e_dec, E, T, xl_d, xr_d, att_d,
                                              escr, mbuf);
  dec_norm<<<cdiv(T, 256), 256, 0, stream>>>(e_dec, E, T, escr, mbuf, ascr, sbuf);
  dec_agg<<<cdiv(T, 256), 256, 0, stream>>>(e_dec, E, T, ascr, sbuf, xl_d, lacc);
  decide<<<cdiv(N, 256), 256, 0, stream>>>(lacc, bias_d, gumbel, gates, N);

  // ---- gated shared GATs (edge_0 -> gate 0, edge_1 -> gate 1) -------------
  const long long* esets[2] = {e_0, e_1};
  for (int which = 0; which < 2; ++which) {
    const long long* ei = esets[which];
    fill_f32<<<cdiv(N, 256), 256, 0, stream>>>(mbuf, NEG_INF, (size_t)N);
    fill_f32<<<cdiv(N, 256), 256, 0, stream>>>(sbuf, 0.0f, (size_t)N);
    gat_score<<<cdiv(T, 8), 256, 0, stream>>>(ei, E, T, xl_g, xr_g, att_g,
                                              gates, which, escr, mbuf);
    gat_norm<<<cdiv(T, 256), 256, 0, stream>>>(ei, E, T, escr, mbuf,
                                               gates, which, ascr, sbuf);
    gat_agg<<<cdiv(T, 8), 256, 0, stream>>>(ei, E, T, ascr, sbuf, xl_g,
                                            gates, which, out);
  }

  // gates sum to 1 per node -> bias added once
  add_bias128<<<cdiv(NO, 256), 256, 0, stream>>>(out, bias_g, NO);
}